// MAB_9294309228645
// MI455X (gfx1250) — compile-verified
//
#include <hip/hip_runtime.h>

namespace {

constexpr int Bn  = 8;
constexpr int Ln  = 1024;
constexpr int Dn  = 512;
constexpr int Hn  = 8;
constexpr float SCALE = 0.04419417382415922f; // 1/sqrt(512)  (ref scales by sqrt(DV))
constexpr float EPSf  = 1e-5f;

typedef __attribute__((ext_vector_type(2))) float v2f;
typedef __attribute__((ext_vector_type(8))) float v8f;

__device__ __forceinline__ v8f wmma4(v2f a, v2f b, v8f c) {
  // D = A(16x4,f32) * B(4x16,f32) + C(16x16,f32)
  return __builtin_amdgcn_wmma_f32_16x16x4_f32(
      /*neg_a=*/false, a, /*neg_b=*/false, b,
      /*c_mod=*/(short)0, c, /*reuse_a=*/false, /*reuse_b=*/false);
}

// ---------------------------------------------------------------------------
// Y[M x 512] = X[M x 512] @ W[512 x 512] + bias
// EPI==1:   Y = res + relu(X@W + bias)
// block = 128 threads (4 waves), block tile 64x64, wave tile 16x64.
// ---------------------------------------------------------------------------
template <int EPI>
__global__ __launch_bounds__(128) void gemm512_kernel(
    const float* __restrict__ X, const float* __restrict__ W,
    const float* __restrict__ bias, const float* __restrict__ res,
    float* __restrict__ Y) {
  __shared__ float Xs[64][36];  // 64 rows x 32 k (pad 4 -> stride 36)
  __shared__ float Ws[32][68];  // 32 k x 64 n (pad 4 -> stride 68)

  const int tid  = threadIdx.x;
  const int lane = tid & 31;
  const int wave = tid >> 5;
  const int m0 = blockIdx.x * 64;
  const int n0 = blockIdx.y * 64;

  const int mrow = lane & 15;        // M (and N) index within a 16-tile
  const int klo  = (lane >> 4) * 2;  // K offset owned by this lane-half
  const int ml   = wave * 16 + mrow;

  v8f acc[4] = {};

  for (int k0 = 0; k0 < Dn; k0 += 32) {
#pragma unroll
    for (int i = 0; i < 4; ++i) {           // 64x32 X block: 512 float4 slots
      const int idx = tid + i * 128;
      const int r = idx >> 3, c4 = idx & 7;
      *(float4*)&Xs[r][c4 * 4] =
          *(const float4*)(X + (size_t)(m0 + r) * Dn + k0 + c4 * 4);
    }
#pragma unroll
    for (int i = 0; i < 4; ++i) {           // 32x64 W block
      const int idx = tid + i * 128;
      const int r = idx >> 4, c4 = idx & 15;
      *(float4*)&Ws[r][c4 * 4] =
          *(const float4*)(W + (size_t)(k0 + r) * Dn + n0 + c4 * 4);
    }
    __syncthreads();

#pragma unroll
    for (int kk = 0; kk < 32; kk += 4) {
      v2f a;
      a.x = Xs[ml][kk + klo];
      a.y = Xs[ml][kk + klo + 1];
#pragma unroll
      for (int nt = 0; nt < 4; ++nt) {
        v2f b;
        b.x = Ws[kk + klo][nt * 16 + mrow];
        b.y = Ws[kk + klo + 1][nt * 16 + mrow];
        acc[nt] = wmma4(a, b, acc[nt]);
      }
    }
    __syncthreads();
  }

  const int rbase = m0 + wave * 16 + ((lane >> 4) << 3);
#pragma unroll
  for (int nt = 0; nt < 4; ++nt) {
    const int col = n0 + nt * 16 + mrow;
    const float bv = bias[col];
#pragma unroll
    for (int v = 0; v < 8; ++v) {
      const size_t off = (size_t)(rbase + v) * Dn + col;
      float val = acc[nt][v] + bv;
      if (EPI == 1) val = res[off] + fmaxf(val, 0.0f);
      Y[off] = val;
    }
  }
}

// ---------------------------------------------------------------------------
// colsum[b,h,k] = sum_q exp(S[q,k]*SCALE)   (query-axis softmax denominator)
// grid: (ktile=16, qtile=16, bh=64), block 128
// ---------------------------------------------------------------------------
__global__ __launch_bounds__(128) void colsum_kernel(
    const float* __restrict__ Qp, const float* __restrict__ Kp,
    float* __restrict__ colsum) {
  __shared__ float Qs[64][68];
  __shared__ float Ks[64][68];
  __shared__ float cpart[64];

  const int tid  = threadIdx.x;
  const int lane = tid & 31;
  const int wave = tid >> 5;
  const int bh = blockIdx.z, b = bh >> 3, h = bh & 7;
  const int q0 = blockIdx.y * 64, k0c = blockIdx.x * 64;

  const float* Qbase = Qp + (size_t)b * Ln * Dn + h * 64;
  const float* Kbase = Kp + (size_t)b * Ln * Dn + h * 64;

#pragma unroll
  for (int i = 0; i < 8; ++i) {  // 64x64 tiles, 1024 float4 slots each
    const int idx = tid + i * 128;
    const int r = idx >> 4, c4 = idx & 15;
    *(float4*)&Qs[r][c4 * 4] =
        *(const float4*)(Qbase + (size_t)(q0 + r) * Dn + c4 * 4);
    *(float4*)&Ks[r][c4 * 4] =
        *(const float4*)(Kbase + (size_t)(k0c + r) * Dn + c4 * 4);
  }
  if (tid < 64) cpart[tid] = 0.0f;
  __syncthreads();

  const int mrow = lane & 15;
  const int klo  = (lane >> 4) * 2;
  const int ml   = wave * 16 + mrow;

  v8f acc[4] = {};
#pragma unroll
  for (int kk = 0; kk < 64; kk += 4) {
    v2f a;
    a.x = Qs[ml][kk + klo];
    a.y = Qs[ml][kk + klo + 1];
#pragma unroll
    for (int nt = 0; nt < 4; ++nt) {
      v2f bb;  // B[d][k] = Kh[k][d] : transposed LDS read (stride-68, bank-clean)
      bb.x = Ks[nt * 16 + mrow][kk + klo];
      bb.y = Ks[nt * 16 + mrow][kk + klo + 1];
      acc[nt] = wmma4(a, bb, acc[nt]);
    }
  }

#pragma unroll
  for (int nt = 0; nt < 4; ++nt) {
    float s = 0.0f;
#pragma unroll
    for (int v = 0; v < 8; ++v) s += __expf(acc[nt][v] * SCALE);
    s += __shfl_xor(s, 16);              // combine M and M+8 halves
    if (lane < 16) atomicAdd(&cpart[nt * 16 + lane], s);  // ds_add_f32
  }
  __syncthreads();
  if (tid < 64)
    atomicAdd(&colsum[(size_t)bh * Ln + k0c + tid], cpart[tid]);  // global f32 atomic
}

// ---------------------------------------------------------------------------
// attn[b,q,h*64+d] = Qh + sum_k (exp(S*SCALE)/colsum[k]) * Vh[k,d]
// grid: (qtile=16, bh=64), block 128.  Ks is reused as the P staging buffer.
// ---------------------------------------------------------------------------
__global__ __launch_bounds__(128) void attn_kernel(
    const float* __restrict__ Qp, const float* __restrict__ Kp,
    const float* __restrict__ Vp, const float* __restrict__ colsum,
    float* __restrict__ Yout) {
  __shared__ float Qs[64][68];
  __shared__ float Ks[64][68];  // K tile, then reused for P tile
  __shared__ float Vs[64][68];

  const int tid  = threadIdx.x;
  const int lane = tid & 31;
  const int wave = tid >> 5;
  const int bh = blockIdx.y, b = bh >> 3, h = bh & 7;
  const int q0 = blockIdx.x * 64;

  const float* Qbase = Qp + (size_t)b * Ln * Dn + h * 64;
  const float* Kbase = Kp + (size_t)b * Ln * Dn + h * 64;
  const float* Vbase = Vp + (size_t)b * Ln * Dn + h * 64;

#pragma unroll
  for (int i = 0; i < 8; ++i) {
    const int idx = tid + i * 128;
    const int r = idx >> 4, c4 = idx & 15;
    *(float4*)&Qs[r][c4 * 4] =
        *(const float4*)(Qbase + (size_t)(q0 + r) * Dn + c4 * 4);
  }
  __syncthreads();

  const int mrow = lane & 15;
  const int klo  = (lane >> 4) * 2;
  const int ml   = wave * 16 + mrow;
  const int r0   = wave * 16 + ((lane >> 4) << 3);

  v8f accO[4] = {};

  for (int kt = 0; kt < 16; ++kt) {
    const int k0c = kt * 64;
#pragma unroll
    for (int i = 0; i < 8; ++i) {
      const int idx = tid + i * 128;
      const int r = idx >> 4, c4 = idx & 15;
      *(float4*)&Ks[r][c4 * 4] =
          *(const float4*)(Kbase + (size_t)(k0c + r) * Dn + c4 * 4);
      *(float4*)&Vs[r][c4 * 4] =
          *(const float4*)(Vbase + (size_t)(k0c + r) * Dn + c4 * 4);
    }
    __syncthreads();

    // S tile = Qh @ Kh^T
    v8f accS[4] = {};
#pragma unroll
    for (int kk = 0; kk < 64; kk += 4) {
      v2f a;
      a.x = Qs[ml][kk + klo];
      a.y = Qs[ml][kk + klo + 1];
#pragma unroll
      for (int nt = 0; nt < 4; ++nt) {
        v2f bb;
        bb.x = Ks[nt * 16 + mrow][kk + klo];
        bb.y = Ks[nt * 16 + mrow][kk + klo + 1];
        accS[nt] = wmma4(a, bb, accS[nt]);
      }
    }
    __syncthreads();  // everyone done reading Ks before it becomes P

    // P = exp(S*SCALE) / colsum[k] -> stage into Ks
#pragma unroll
    for (int nt = 0; nt < 4; ++nt) {
      const int kc = k0c + nt * 16 + mrow;
      const float rcs = 1.0f / colsum[(size_t)bh * Ln + kc];
#pragma unroll
      for (int v = 0; v < 8; ++v)
        Ks[r0 + v][nt * 16 + mrow] = __expf(accS[nt][v] * SCALE) * rcs;
    }
    __syncthreads();

    // accO += P @ Vh
#pragma unroll
    for (int kk = 0; kk < 64; kk += 4) {
      v2f a;
      a.x = Ks[ml][kk + klo];
      a.y = Ks[ml][kk + klo + 1];
#pragma unroll
      for (int nt = 0; nt < 4; ++nt) {
        v2f bb;
        bb.x = Vs[kk + klo][nt * 16 + mrow];
        bb.y = Vs[kk + klo + 1][nt * 16 + mrow];
        accO[nt] = wmma4(a, bb, accO[nt]);
      }
    }
    __syncthreads();
  }

  // residual (+Qh) and store merged-head layout
  float* Ybase = Yout + (size_t)b * Ln * Dn + h * 64;
#pragma unroll
  for (int nt = 0; nt < 4; ++nt) {
    const int d = nt * 16 + mrow;
#pragma unroll
    for (int v = 0; v < 8; ++v) {
      const int q = q0 + r0 + v;
      Ybase[(size_t)q * Dn + d] = Qs[r0 + v][d] + accO[nt][v];
    }
  }
}

// ---------------------------------------------------------------------------
// Row LayerNorm over 512: one wave per row, 8 rows per 256-thread block.
// ---------------------------------------------------------------------------
__global__ __launch_bounds__(256) void ln_kernel(
    const float* __restrict__ X, const float* __restrict__ gam,
    const float* __restrict__ bet, float* __restrict__ Y) {
  const int lane = threadIdx.x & 31;
  const int wave = threadIdx.x >> 5;
  const size_t row = (size_t)blockIdx.x * 8 + wave;
  const float* x = X + row * Dn + lane * 16;

  float4 v[4];
#pragma unroll
  for (int i = 0; i < 4; ++i) v[i] = *(const float4*)(x + i * 4);

  float s = 0.0f;
#pragma unroll
  for (int i = 0; i < 4; ++i)
#pragma unroll
    for (int j = 0; j < 4; ++j) s += ((const float*)&v[i])[j];
#pragma unroll
  for (int off = 16; off; off >>= 1) s += __shfl_xor(s, off);
  const float mu = s * (1.0f / Dn);

  float vs = 0.0f;
#pragma unroll
  for (int i = 0; i < 4; ++i)
#pragma unroll
    for (int j = 0; j < 4; ++j) {
      const float d = ((const float*)&v[i])[j] - mu;
      vs += d * d;
    }
#pragma unroll
  for (int off = 16; off; off >>= 1) vs += __shfl_xor(vs, off);
  const float inv = rsqrtf(vs * (1.0f / Dn) + EPSf);

  const int cb = lane * 16;
  float* y = Y + row * Dn + cb;
#pragma unroll
  for (int i = 0; i < 4; ++i) {
    float4 o;
#pragma unroll
    for (int j = 0; j < 4; ++j) {
      const int c = cb + i * 4 + j;
      ((float*)&o)[j] =
          (((const float*)&v[i])[j] - mu) * inv * gam[c] + bet[c];
    }
    *(float4*)(y + i * 4) = o;
  }
}

__global__ void zero_kernel(float* __restrict__ p, int n) {
  const int i = blockIdx.x * blockDim.x + threadIdx.x;
  if (i < n) p[i] = 0.0f;
}

}  // namespace

extern "C" void kernel_launch(void* const* d_in, const int* in_sizes, int n_in,
                              void* d_out, int out_size, void* d_ws,
                              size_t ws_size, hipStream_t stream) {
  (void)in_sizes; (void)n_in; (void)out_size; (void)ws_size;
  const float* Q  = (const float*)d_in[0];
  const float* K  = (const float*)d_in[1];
  const float* Wq = (const float*)d_in[2];
  const float* bq = (const float*)d_in[3];
  const float* Wk = (const float*)d_in[4];
  const float* bk = (const float*)d_in[5];
  const float* Wv = (const float*)d_in[6];
  const float* bv = (const float*)d_in[7];
  const float* Wo = (const float*)d_in[8];
  const float* bo = (const float*)d_in[9];
  const float* g0 = (const float*)d_in[10];
  const float* b0 = (const float*)d_in[11];
  const float* g1 = (const float*)d_in[12];
  const float* b1 = (const float*)d_in[13];
  float* out = (float*)d_out;

  float* ws = (float*)d_ws;
  const size_t MN = (size_t)Bn * Ln * Dn;  // 4M floats
  float* Qp   = ws;
  float* Kp   = ws + MN;
  float* Vp   = ws + 2 * MN;
  float* attn = ws + 3 * MN;
  float* cs   = ws + 4 * MN;   // 64K floats
  float* X1 = Qp;              // Qp dead after attention
  float* X2 = Kp;              // Kp dead after attention

  const int M = Bn * Ln;  // 8192
  const dim3 gGemm(M / 64, Dn / 64);

  gemm512_kernel<0><<<gGemm, 128, 0, stream>>>(Q, Wq, bq, nullptr, Qp);
  gemm512_kernel<0><<<gGemm, 128, 0, stream>>>(K, Wk, bk, nullptr, Kp);
  gemm512_kernel<0><<<gGemm, 128, 0, stream>>>(K, Wv, bv, nullptr, Vp);

  const int csN = Bn * Hn * Ln;  // 65536
  zero_kernel<<<csN / 256, 256, 0, stream>>>(cs, csN);
  colsum_kernel<<<dim3(16, 16, Bn * Hn), 128, 0, stream>>>(Qp, Kp, cs);
  attn_kernel<<<dim3(16, Bn * Hn), 128, 0, stream>>>(Qp, Kp, Vp, cs, attn);

  ln_kernel<<<M / 8, 256, 0, stream>>>(attn, g0, b0, X1);
  gemm512_kernel<1><<<gGemm, 128, 0, stream>>>(X1, Wo, bo, X1, X2);
  ln_kernel<<<M / 8, 256, 0, stream>>>(X2, g1, b1, out);
}